// EmbeddingBagCollection_45226005627575
// MI455X (gfx1250) — compile-verified
//
#include <hip/hip_runtime.h>

// Problem constants (from the reference):
#define NUM_T 4
#define VOCAB 1000000
#define DIM   64
#define BAGS  4096      // power of two: 2^12
#define NIDX  81920

#define WPB   8         // waves per block (256 threads)
#define DEPTH 16        // async rows in flight per wave; s_wait uses DEPTH-1 = 15

typedef float v2f __attribute__((ext_vector_type(2)));   // native vector: OK for NT builtins

// One wave32 per bag. Lane l owns dims [2l, 2l+1] (8 bytes) => one
// global_load_async_to_lds_b64 moves an entire 256B embedding row per wave.
__global__ __launch_bounds__(256)
void ebc_pool_kernel(const float* __restrict__ tables,
                     const int*   __restrict__ indices,
                     const int*   __restrict__ offsets,
                     float*       __restrict__ out)
{
    __shared__ float buf[WPB * DEPTH * DIM];   // 32 KB / block

    const int lane = threadIdx.x & 31;
    const int wib  = threadIdx.x >> 5;
    const int gw   = blockIdx.x * WPB + wib;   // bag id in [0, T*B)
    const int t    = gw >> 12;                 // BAGS == 4096
    const int b    = gw & (BAGS - 1);

    const int start = offsets[t * BAGS + b];
    const int end   = (b + 1 < BAGS) ? offsets[t * BAGS + b + 1] : NIDX;
    const int cnt   = end - start;             // wave-uniform; EXEC stays full

    const int*   __restrict__ idxp = indices + (size_t)t * NIDX + start;
    const float* __restrict__ tbl  = tables + (size_t)t * (size_t)VOCAB * DIM;

    // Pull the bag's index stream toward the WGP once, ahead of the pipeline.
    __builtin_prefetch(idxp + DEPTH, 0, 0);    // -> global_prefetch_b8

    // This lane's 8-byte slice within each of this wave's DEPTH LDS slots.
    float* slotBase = buf + wib * (DEPTH * DIM) + lane * 2;

    float ax = 0.f, ay = 0.f;

    // ---- prologue: fill the async pipeline ----
    const int pre = (cnt < DEPTH) ? cnt : DEPTH;
    for (int h = 0; h < pre; ++h) {
        const int idx = idxp[h];
        unsigned long long ga =
            (unsigned long long)(const void*)(tbl + (size_t)idx * DIM + lane * 2);
        unsigned ldsa = (unsigned)(size_t)(const void*)(slotBase + h * DIM);
        asm volatile("global_load_async_to_lds_b64 %0, %1, off"
                     :: "v"(ldsa), "v"(ga) : "memory");
    }

    // ---- steady state: wait oldest, accumulate, refill slot ----
    int i = 0;
    for (; i < cnt - DEPTH; ++i) {
        const int nidx = idxp[i + DEPTH];                 // independent of the wait
        // async loads complete in order: <=15 outstanding => row i has landed
        asm volatile("s_wait_asynccnt 15" ::: "memory");
        const v2f v = *(const v2f*)(slotBase + (i & (DEPTH - 1)) * DIM);
        ax += v.x; ay += v.y;
        unsigned long long ga =
            (unsigned long long)(const void*)(tbl + (size_t)nidx * DIM + lane * 2);
        unsigned ldsa =
            (unsigned)(size_t)(const void*)(slotBase + (i & (DEPTH - 1)) * DIM);
        asm volatile("global_load_async_to_lds_b64 %0, %1, off"
                     :: "v"(ldsa), "v"(ga) : "memory");
    }

    // ---- drain ----
    asm volatile("s_wait_asynccnt 0" ::: "memory");
    for (; i < cnt; ++i) {
        const v2f v = *(const v2f*)(slotBase + (i & (DEPTH - 1)) * DIM);
        ax += v.x; ay += v.y;
    }

    v2f r; r.x = ax; r.y = ay;                    // empty bags store zeros
    // Output is write-once: NT store keeps the 192MB L2 for table-row reuse.
    __builtin_nontemporal_store(r, (v2f*)(out + (size_t)gw * DIM + lane * 2));
}

extern "C" void kernel_launch(void* const* d_in, const int* in_sizes, int n_in,
                              void* d_out, int out_size, void* d_ws, size_t ws_size,
                              hipStream_t stream) {
    (void)in_sizes; (void)n_in; (void)d_ws; (void)ws_size; (void)out_size;
    const float* tables  = (const float*)d_in[0];
    const int*   indices = (const int*)d_in[1];
    const int*   offsets = (const int*)d_in[2];
    float*       out     = (float*)d_out;

    const int totalBags = NUM_T * BAGS;            // 16384 waves, 1 per bag
    const int blocks    = totalBags / WPB;         // 2048 blocks of 256 threads
    ebc_pool_kernel<<<blocks, 256, 0, stream>>>(tables, indices, offsets, out);
}